// BrainBottleneckLocal_8409545966415
// MI455X (gfx1250) — compile-verified
//
#include <hip/hip_runtime.h>
#include <hip/hip_bf16.h>

// MI455X (gfx1250) implementation. fp32 end-to-end using V_WMMA_F32_16X16X4_F32:
// the workload is HBM-bound on the 604MB fp32 locally-connected weight tensor,
// so native-f32 matrix ops (no conversion traffic) are the roofline-optimal path.
// The LC weight stream is staged into LDS with GLOBAL_LOAD_ASYNC_TO_LDS_B32
// (ASYNCcnt) and double-buffered so the copy overlaps the WMMA inner loop.

typedef __attribute__((ext_vector_type(2))) float v2f;
typedef __attribute__((ext_vector_type(8))) float v8f;

#define EPSBN 1e-5f

#define NB    64     // batch
#define C_IN  1024
#define HW    256    // 16x16
#define WIDTH 256
#define C_OUT 1024

static __device__ __forceinline__ v8f wmma_f32(v2f a, v2f b, v8f c) {
  // 8 args: (neg_a, A, neg_b, B, c_mod, C, reuse_a, reuse_b)
  return __builtin_amdgcn_wmma_f32_16x16x4_f32(false, a, false, b, (short)0, c,
                                               false, false);
}

// Async copy one dword global -> LDS (ASYNCcnt). LDS dest = low 32 bits of the
// generic pointer (flat LDS aperture keeps the LDS byte offset in addr[31:0]).
static __device__ __forceinline__ void async_g2l_b32(void* lds_dst,
                                                     const void* gsrc) {
  unsigned dst = (unsigned)(unsigned long long)(uintptr_t)lds_dst;
  unsigned long long ga = (unsigned long long)(uintptr_t)gsrc;
  asm volatile("global_load_async_to_lds_b32 %0, %1, off"
               :: "v"(dst), "v"(ga) : "memory");
}
static __device__ __forceinline__ void wait_asynccnt0() {
  asm volatile("s_wait_asynccnt 0x0" ::: "memory");
}

// ---------------------------------------------------------------------------
// Kernel 1: y1[n][o][p] = relu(bn1(sum_c x[n][c][p] * w1[o][c]))
// Per-wave tile: 16 (p) x 64 (o), K=1024 stepped by 4.
// ---------------------------------------------------------------------------
__global__ __launch_bounds__(256) void conv1_bn_relu(
    const float* __restrict__ x, const float* __restrict__ w1,
    const float* __restrict__ bn_g, const float* __restrict__ bn_b,
    const float* __restrict__ bn_m, const float* __restrict__ bn_v,
    float* __restrict__ y1) {
  const int tid  = threadIdx.x;
  const int lane = tid & 31;
  const int wave = tid >> 5;
  const int wid  = blockIdx.x * 8 + wave;     // 4096 waves total
  const int n    = wid >> 6;                  // 0..63
  const int rem  = wid & 63;
  const int m0   = (rem >> 2) * 16;           // p tile base (16 tiles)
  const int o0   = (rem & 3) * 64;            // o tile base (4 tiles)
  const int l16  = lane & 15;
  const int hi   = lane >> 4;                 // 0 or 1
  const int koff = hi << 1;

  const float* xa = x + (size_t)n * C_IN * HW;
  v8f acc[4] = {};

  #pragma unroll 4
  for (int kb = 0; kb < C_IN; kb += 4) {
    v2f a;
    a.x = xa[(size_t)(kb + koff) * HW + m0 + l16];
    a.y = xa[(size_t)(kb + koff + 1) * HW + m0 + l16];
    const float* wb = w1 + (size_t)(o0 + l16) * C_IN + kb + koff;
    if ((kb & 63) == 0) __builtin_prefetch(xa + (size_t)(kb + 64) * HW + m0, 0, 0);
    #pragma unroll
    for (int j = 0; j < 4; ++j) {
      v2f b = *(const v2f*)(wb + (size_t)j * 16 * C_IN);  // (c, c+1) contiguous
      acc[j] = wmma_f32(a, b, acc[j]);
    }
  }

  float* yb = y1 + (size_t)n * WIDTH * HW;
  #pragma unroll
  for (int j = 0; j < 4; ++j) {
    const int o  = o0 + j * 16 + l16;
    const float sc = bn_g[o] * rsqrtf(bn_v[o] + EPSBN);
    const float sh = bn_b[o] - bn_m[o] * sc;
    #pragma unroll
    for (int r = 0; r < 8; ++r) {
      const int p = m0 + r + (hi << 3);
      yb[(size_t)o * HW + p] = fmaxf(acc[j][r] * sc + sh, 0.f);
    }
  }
}

// ---------------------------------------------------------------------------
// Kernel 2: locally connected 3x3 + BN + ReLU.
// Per block: one spatial position p, 128-wide o tile; GEMM M=64(n) x K=2304(c,kk)
// x N=128(o). K chunked 4 channels (Kt=36); weight tile async-copied to LDS
// (double buffered, overlapped with WMMA); activation patches staged with
// ds_store (zero-padded border).
// ---------------------------------------------------------------------------
#define LC_CC 4
#define LC_KT 36
__global__ __launch_bounds__(256) void lc_bn_relu(
    const float* __restrict__ y1, const float* __restrict__ lcw,
    const float* __restrict__ bn_g, const float* __restrict__ bn_b,
    const float* __restrict__ bn_m, const float* __restrict__ bn_v,
    float* __restrict__ y2) {
  __shared__ float Bl[2][LC_KT][136];  // 128 o cols, padded stride (banks)
  __shared__ float Al[2][LC_KT][72];   // 64 n cols, padded stride

  const int p    = blockIdx.x;            // 0..255
  const int o0   = blockIdx.y * 128;      // 0 or 128
  const int tid  = threadIdx.x;
  const int lane = tid & 31;
  const int wave = tid >> 5;
  const int l16  = lane & 15;
  const int hi   = lane >> 4;
  const int koff = hi << 1;
  const int m0   = (wave & 3) * 16;       // n-batch tile base
  const int n0   = (wave >> 2) * 64;      // local o base (0 or 64)
  const int h = p >> 4, w = p & 15;

  // Staging-role constants for this thread.
  const int b_oo = tid >> 2;              // 0..63 (x2 via i loop -> 0..127)
  const int b_cc = tid & 3;
  const int a_nn = tid >> 2;              // 0..63
  const int a_cc = tid & 3;

  v8f acc[4] = {};

  // ---- tile staging helpers (inlined twice: prologue + loop) ----
  auto issueB = [&](int c0, int buf) {
    #pragma unroll
    for (int i = 0; i < 2; ++i) {
      const int oo = b_oo + i * 64;       // 0..127
      const float* src = lcw +
          ((size_t)(o0 + oo) * WIDTH + (c0 + b_cc)) * (size_t)(HW * 9) +
          (size_t)p * 9;
      #pragma unroll
      for (int kk = 0; kk < 9; ++kk)
        async_g2l_b32(&Bl[buf][b_cc * 9 + kk][oo], src + kk);
    }
  };
  auto stageA = [&](int c0, int buf) {
    const float* srcb =
        y1 + (size_t)a_nn * WIDTH * HW + (size_t)(c0 + a_cc) * HW;
    #pragma unroll
    for (int kk = 0; kk < 9; ++kk) {
      const int hh = h + (kk / 3) - 1;
      const int ww = w + (kk % 3) - 1;
      float v = 0.f;
      if (hh >= 0 && hh < 16 && ww >= 0 && ww < 16) v = srcb[hh * 16 + ww];
      Al[buf][a_cc * 9 + kk][a_nn] = v;
    }
  };

  // Prologue: fill buffer 0.
  issueB(0, 0);
  stageA(0, 0);
  wait_asynccnt0();
  __syncthreads();

  for (int c0 = 0; c0 < WIDTH; c0 += LC_CC) {
    const int cur = (c0 / LC_CC) & 1;
    if (c0 + LC_CC < WIDTH) {           // overlap next tile with compute
      issueB(c0 + LC_CC, cur ^ 1);
      stageA(c0 + LC_CC, cur ^ 1);
    }

    #pragma unroll
    for (int kb = 0; kb < LC_KT; kb += 4) {
      v2f a;
      a.x = Al[cur][kb + koff][m0 + l16];
      a.y = Al[cur][kb + koff + 1][m0 + l16];
      #pragma unroll
      for (int j = 0; j < 4; ++j) {
        v2f b;
        b.x = Bl[cur][kb + koff][n0 + j * 16 + l16];
        b.y = Bl[cur][kb + koff + 1][n0 + j * 16 + l16];
        acc[j] = wmma_f32(a, b, acc[j]);
      }
    }

    wait_asynccnt0();   // this wave's async copies for the next buffer landed
    __syncthreads();    // publish LDS writes to all waves
  }

  #pragma unroll
  for (int j = 0; j < 4; ++j) {
    const int o  = o0 + n0 + j * 16 + l16;
    const float sc = bn_g[o] * rsqrtf(bn_v[o] + EPSBN);
    const float sh = bn_b[o] - bn_m[o] * sc;
    #pragma unroll
    for (int r = 0; r < 8; ++r) {
      const int nn = m0 + r + (hi << 3);
      y2[(size_t)nn * WIDTH * HW + (size_t)o * HW + p] =
          fmaxf(acc[j][r] * sc + sh, 0.f);
    }
  }
}

// ---------------------------------------------------------------------------
// Kernel 3: y3 = relu(bn3(y2 @ w3^T) + x)   (K=256, N=1024)
// ---------------------------------------------------------------------------
__global__ __launch_bounds__(256) void conv3_bn_add_relu(
    const float* __restrict__ y2, const float* __restrict__ w3,
    const float* __restrict__ bn_g, const float* __restrict__ bn_b,
    const float* __restrict__ bn_m, const float* __restrict__ bn_v,
    const float* __restrict__ x, float* __restrict__ y3) {
  const int tid  = threadIdx.x;
  const int lane = tid & 31;
  const int wave = tid >> 5;
  const int wid  = blockIdx.x * 8 + wave;    // 16384 waves
  const int n    = wid >> 8;
  const int rem  = wid & 255;
  const int m0   = (rem >> 4) * 16;          // p tile
  const int o0   = (rem & 15) * 64;          // o tile
  const int l16  = lane & 15;
  const int hi   = lane >> 4;
  const int koff = hi << 1;

  const float* ab = y2 + (size_t)n * WIDTH * HW;
  v8f acc[4] = {};

  #pragma unroll 4
  for (int kb = 0; kb < WIDTH; kb += 4) {
    v2f a;
    a.x = ab[(size_t)(kb + koff) * HW + m0 + l16];
    a.y = ab[(size_t)(kb + koff + 1) * HW + m0 + l16];
    const float* wb = w3 + (size_t)(o0 + l16) * WIDTH + kb + koff;
    #pragma unroll
    for (int j = 0; j < 4; ++j) {
      v2f b = *(const v2f*)(wb + (size_t)j * 16 * WIDTH);
      acc[j] = wmma_f32(a, b, acc[j]);
    }
  }

  const float* xb = x + (size_t)n * C_OUT * HW;
  float* yb = y3 + (size_t)n * C_OUT * HW;
  #pragma unroll
  for (int j = 0; j < 4; ++j) {
    const int o  = o0 + j * 16 + l16;
    const float sc = bn_g[o] * rsqrtf(bn_v[o] + EPSBN);
    const float sh = bn_b[o] - bn_m[o] * sc;
    #pragma unroll
    for (int r = 0; r < 8; ++r) {
      const int p = m0 + r + (hi << 3);
      const float idn = xb[(size_t)o * HW + p];
      yb[(size_t)o * HW + p] = fmaxf(acc[j][r] * sc + sh + idn, 0.f);
    }
  }
}

// ---------------------------------------------------------------------------
// Kernel 4a: build inhibition matrix g[o][c] (column-wise Gaussian softmax;
// the 1/(2.5066*sigma) prefactor cancels in the column normalization).
// One block per column c.
// ---------------------------------------------------------------------------
__global__ __launch_bounds__(256) void build_g(const float* __restrict__ sigmas,
                                               float* __restrict__ g) {
  const int c = blockIdx.x;
  const int t = threadIdx.x;
  __shared__ float red[256];

  const float sig    = fmaxf(sigmas[c], 0.5f);
  const float inv2s2 = 1.0f / (2.0f * sig * sig);

  float vals[4];
  float s = 0.f;
  #pragma unroll
  for (int i = 0; i < 4; ++i) {
    const int o = t + i * 256;
    const int j = (c - o) & (C_OUT - 1);             // (c - o) mod 1024
    const float d = fabsf((float)(j + 1) - 513.0f);  // ci[(c-o)%C]
    const float v = __expf(-d * d * inv2s2);
    vals[i] = v;
    s += v;
  }
  red[t] = s;
  __syncthreads();
  for (int off = 128; off > 0; off >>= 1) {
    if (t < off) red[t] += red[t + off];
    __syncthreads();
  }
  const float inv = 1.0f / red[0];
  #pragma unroll
  for (int i = 0; i < 4; ++i)
    g[(size_t)(t + i * 256) * C_OUT + c] = vals[i] * inv;
}

// ---------------------------------------------------------------------------
// Kernel 4b: inh = y3 @ g^T (K=1024, N=1024); out = y3 / (inh + 1)
// ---------------------------------------------------------------------------
__global__ __launch_bounds__(256) void inhibit(
    const float* __restrict__ y3, const float* __restrict__ g,
    float* __restrict__ out) {
  const int tid  = threadIdx.x;
  const int lane = tid & 31;
  const int wave = tid >> 5;
  const int wid  = blockIdx.x * 8 + wave;    // 16384 waves
  const int n    = wid >> 8;
  const int rem  = wid & 255;
  const int m0   = (rem >> 4) * 16;
  const int o0   = (rem & 15) * 64;
  const int l16  = lane & 15;
  const int hi   = lane >> 4;
  const int koff = hi << 1;

  const float* ab = y3 + (size_t)n * C_OUT * HW;
  v8f acc[4] = {};

  #pragma unroll 4
  for (int kb = 0; kb < C_OUT; kb += 4) {
    v2f a;
    a.x = ab[(size_t)(kb + koff) * HW + m0 + l16];
    a.y = ab[(size_t)(kb + koff + 1) * HW + m0 + l16];
    const float* gb = g + (size_t)(o0 + l16) * C_OUT + kb + koff;
    if ((kb & 63) == 0) __builtin_prefetch(ab + (size_t)(kb + 64) * HW + m0, 0, 0);
    #pragma unroll
    for (int j = 0; j < 4; ++j) {
      v2f b = *(const v2f*)(gb + (size_t)j * 16 * C_OUT);
      acc[j] = wmma_f32(a, b, acc[j]);
    }
  }

  float* ob = out + (size_t)n * C_OUT * HW;
  #pragma unroll
  for (int j = 0; j < 4; ++j) {
    const int o = o0 + j * 16 + l16;
    #pragma unroll
    for (int r = 0; r < 8; ++r) {
      const int p = m0 + r + (hi << 3);
      const float num = ab[(size_t)o * HW + p];
      ob[(size_t)o * HW + p] = num / (acc[j][r] + 1.0f);
    }
  }
}

// ---------------------------------------------------------------------------
extern "C" void kernel_launch(void* const* d_in, const int* in_sizes, int n_in,
                              void* d_out, int out_size, void* d_ws,
                              size_t ws_size, hipStream_t stream) {
  const float* x   = (const float*)d_in[0];
  const float* w1  = (const float*)d_in[1];
  const float* g1  = (const float*)d_in[2];
  const float* b1  = (const float*)d_in[3];
  const float* m1  = (const float*)d_in[4];
  const float* v1  = (const float*)d_in[5];
  const float* lcw = (const float*)d_in[6];
  const float* g2  = (const float*)d_in[7];
  const float* b2  = (const float*)d_in[8];
  const float* m2  = (const float*)d_in[9];
  const float* v2  = (const float*)d_in[10];
  const float* w3  = (const float*)d_in[11];
  const float* g3  = (const float*)d_in[12];
  const float* b3  = (const float*)d_in[13];
  const float* m3  = (const float*)d_in[14];
  const float* v3  = (const float*)d_in[15];
  const float* sig = (const float*)d_in[16];
  float* out = (float*)d_out;

  char* ws = (char*)d_ws;
  float* y1 = (float*)(ws);                          // 16 MB
  float* y2 = (float*)(ws + (size_t)(16u << 20));    // 16 MB
  float* y3 = (float*)(ws + (size_t)(32u << 20));    // 64 MB
  float* gm = (float*)(ws + (size_t)(96u << 20));    // 4 MB

  build_g<<<1024, 256, 0, stream>>>(sig, gm);
  conv1_bn_relu<<<512, 256, 0, stream>>>(x, w1, g1, b1, m1, v1, y1);
  lc_bn_relu<<<dim3(256, 2), 256, 0, stream>>>(y1, lcw, g2, b2, m2, v2, y2);
  conv3_bn_add_relu<<<2048, 256, 0, stream>>>(y2, w3, g3, b3, m3, v3, x, y3);
  inhibit<<<2048, 256, 0, stream>>>(y3, gm, out);
}